// PygAttPlus_55516747268137
// MI455X (gfx1250) — compile-verified
//
#include <hip/hip_runtime.h>

#define HEADS 4
#define OUT_CH 32
#define FEAT (HEADS * OUT_CH) /* 128 */

typedef __attribute__((ext_vector_type(2))) float v2f;
typedef __attribute__((ext_vector_type(4))) float v4f;
typedef __attribute__((ext_vector_type(8))) float v8f;

// Head-row GEMM: Xh(R x 32) @ W2(32 x 16), R = 4*n_nodes head-rows.
//   W2 col0 = w_i (w[0:32]), col1 = w_j (w[32:64]), cols 2..15 = 0.
// Output scores: S[node*8 + h]     = dot(x[node,h,:], w_i)
//                S[node*8 + 4 + h] = dot(x[node,h,:], w_j)
// One wave per 16-row tile; 8 chained V_WMMA_F32_16X16X4_F32, K fully dense.
__global__ void __launch_bounds__(32) node_score_wmma(const float* __restrict__ x,
                                                      const float* __restrict__ w,
                                                      float* __restrict__ S,
                                                      int n_rows /* = 4*n_nodes */) {
    const int lane = threadIdx.x;      // one wave32 per block
    const int m0   = blockIdx.x * 16;  // tile head-row base
    const int m    = lane & 15;        // A: row M ; B/C: col N
    const int kh   = lane >> 4;        // K-half (0/1)

    // ---- Preload all B fragments into registers (branch-free) ----
    // B[k][n]: n==0 -> w_i[k], n==1 -> w_j[k], else 0.
    // Unconditional loads (index always in [0,64)), zeroed by select.
    const int  wbase  = (m == 1) ? OUT_CH : 0;
    const bool active = (m < 2);
    v2f bf[8];
#pragma unroll
    for (int t = 0; t < 8; ++t) {
        const int k = 4 * t + 2 * kh;
        const v2f wv = *(const v2f*)(w + wbase + k);
        bf[t].x = active ? wv.x : 0.f;
        bf[t].y = active ? wv.y : 0.f;
    }

    // ---- Main WMMA chain: A = 16x4 f32 tile of head-rows ----
    const float* xrow = x + (size_t)(m0 + m) * OUT_CH; // head-row, 32 floats
    v8f c = {};
#pragma unroll
    for (int t = 0; t < 8; ++t) {
        // A layout: lanes 0..15 = M, VGPR0 = K(2*kh), VGPR1 = K(2*kh+1)
        const v2f a = *(const v2f*)(xrow + 4 * t + 2 * kh);
        c = __builtin_amdgcn_wmma_f32_16x16x4_f32(
            /*neg_a=*/false, a, /*neg_b=*/false, bf[t],
            /*c_mod=*/(short)0, c, /*reuse_a=*/false, /*reuse_b=*/false);
    }

    // ---- Store tail ----
    // C layout: lane<16 -> N=lane, VGPR v -> M=v; lane>=16 -> N=lane-16, M=8+v.
    // For active lane: row r = r0+v (r0 = m0+kh*8, multiple of 8), and
    //   S-index = (r>>2)*8 + col + (r&3) = 2*r0 + col + {0,1,2,3, 8,9,10,11}[v]
    // -> two contiguous float4 groups, 16B-aligned (2*r0*4 is 64B-aligned).
    if (active) {
        const int col = (m == 1) ? 4 : 0;   // s_i column or s_j column
        const int r0  = m0 + kh * 8;
        float* dst = S + (size_t)2 * r0 + col;
        if (m0 + 16 <= n_rows) {
            // Full tile (always taken when n_rows % 16 == 0): vector stores.
            v4f lo, hi;
            lo.x = c[0]; lo.y = c[1]; lo.z = c[2]; lo.w = c[3];
            hi.x = c[4]; hi.y = c[5]; hi.z = c[6]; hi.w = c[7];
            *(v4f*)(dst)     = lo;
            *(v4f*)(dst + 8) = hi;
        } else {
            // Partial last tile: guarded scalar stores.
#pragma unroll
            for (int v = 0; v < 8; ++v) {
                const int r = r0 + v;
                if (r < n_rows) dst[(v >> 2) * 8 + (v & 3)] = c[v];
            }
        }
    }
}

__global__ void zero_out(float4* __restrict__ out, int n4) {
    const int i = blockIdx.x * blockDim.x + threadIdx.x;
    if (i < n4) out[i] = make_float4(0.f, 0.f, 0.f, 0.f);
}

// One wave per edge; lane owns 4 channels (float4), head = lane>>3.
// Edge scalars loaded once by lane 0 and broadcast to SGPRs.
__global__ void __launch_bounds__(256) edge_scatter(const float* __restrict__ x,
                                                    const int* __restrict__ ei,
                                                    const int* __restrict__ ej,
                                                    const float* __restrict__ wts,
                                                    const float* __restrict__ S,
                                                    float* __restrict__ out,
                                                    int n_edges) {
    const int lane = threadIdx.x & 31;
    const int e = blockIdx.x * (blockDim.x >> 5) + (threadIdx.x >> 5);
    if (e >= n_edges) return; // e is wave-uniform: whole wave exits together

    // Wave-uniform edge scalars: load in lane 0, broadcast via readfirstlane.
    int iv = 0, jv = 0;
    unsigned wvb = 0u;
    if (lane == 0) {
        iv  = ei[e];
        jv  = ej[e];
        wvb = __float_as_uint(wts[e]);
    }
    const int   i  = __builtin_amdgcn_readfirstlane(iv);
    const int   j  = __builtin_amdgcn_readfirstlane(jv);
    const float we = __uint_as_float(__builtin_amdgcn_readfirstlane(wvb));

    const int h = lane >> 3;
    const float alpha = S[(size_t)i * 8 + h] + S[(size_t)j * 8 + 4 + h];
    const float beta  = we / (1.f + __expf(-alpha)); // weights * sigmoid(alpha)

    const float4 xj = *(const float4*)(x + (size_t)j * FEAT + lane * 4);
    float* dst = out + (size_t)i * FEAT + lane * 4;
    atomicAdd(dst + 0, xj.x * beta);
    atomicAdd(dst + 1, xj.y * beta);
    atomicAdd(dst + 2, xj.z * beta);
    atomicAdd(dst + 3, xj.w * beta);
}

extern "C" void kernel_launch(void* const* d_in, const int* in_sizes, int n_in,
                              void* d_out, int out_size, void* d_ws, size_t ws_size,
                              hipStream_t stream) {
    const float* x    = (const float*)d_in[0]; // (N_NODES, 128)
    const int*   eidx = (const int*)d_in[1];   // (2, N_EDGES)
    const float* wts  = (const float*)d_in[2]; // (N_EDGES,)
    const float* w    = (const float*)d_in[3]; // (64,)
    float* out = (float*)d_out;                // (N_NODES, 128)
    float* S   = (float*)d_ws;                 // (N_NODES, 8) scores scratch

    const int n_edges = in_sizes[2];
    const int n_nodes = in_sizes[0] / FEAT;
    const int n_rows  = n_nodes * HEADS;
    const int* ei = eidx;
    const int* ej = eidx + n_edges;

    const int n4 = out_size / 4;
    zero_out<<<(n4 + 255) / 256, 256, 0, stream>>>((float4*)out, n4);
    node_score_wmma<<<(n_rows + 15) / 16, 32, 0, stream>>>(x, w, S, n_rows);
    edge_scatter<<<(n_edges + 7) / 8, 256, 0, stream>>>(x, ei, ej, wts, S, out, n_edges);
}